// CapsuleConv2d_6184752906706
// MI455X (gfx1250) — compile-verified
//
#include <hip/hip_runtime.h>
#include <hip/hip_fp16.h>

// CDNA5 / gfx1250 fused CapsuleConv2d + dynamic routing.
//
// Roofline: 17 MB HBM traffic (0.7 us @ 23.3 TB/s) vs ~1.2 GFLOP matrix +
// ~2 GFLOP routing VALU -> compute bound; priors (302 MB if materialized)
// must live in LDS. Priors GEMM uses V_WMMA_F32_16X16X4_F32 (K=4 fp32):
// the capsule reduction is only in_len=8, so two chained K=4 WMMAs give
// 100% K utilization at full fp32 precision (f16 K=32 WMMA would waste 3/4
// of its K dimension here).

typedef __attribute__((ext_vector_type(2))) float v2f;
typedef __attribute__((ext_vector_type(8))) float v8f;

#define N_BATCH   16
#define C_IN      32
#define HW        64
#define F_OUT     4
#define F_IN      4
#define OUT_LEN   8
#define IN_LEN    8
#define TAPS      36          // F_IN * 3 * 3
#define OU        32          // F_OUT * OUT_LEN
#define WAVES     4
#define POS_WAVE  16          // WMMA M dimension
#define POS_BLK   (WAVES * POS_WAVE)
#define LDS_HALFS (WAVES * POS_WAVE * TAPS * OU)   // 73728 halves = 144 KB

__global__ __launch_bounds__(WAVES * 32)
void capsule_routing_wmma(const float* __restrict__ x,
                          const float* __restrict__ w,
                          float* __restrict__ out) {
  extern __shared__ __half pri_lds[];   // [WAVES][16 pos][36 taps][32 (o,u)]

  const int lane = threadIdx.x & 31;
  const int wave = threadIdx.x >> 5;
  const int kgrp = lane >> 4;           // K-half owned by this lane (A & B layout)
  const int m    = lane & 15;           // M index (position) for A loads
  const int nn   = lane & 15;           // N index ((o,u) within half) for B loads

  // prefetch the 36 KB weight tensor into cache (global_prefetch_b8)
  for (int off = threadIdx.x * 256; off < F_OUT * F_IN * 9 * 64 * 4; off += WAVES * 32 * 256)
    __builtin_prefetch((const char*)w + off, 0, 3);

  // ---- position owned by this lane for the A tile --------------------------
  const int P  = blockIdx.x * POS_BLK + wave * POS_WAVE + m;   // < 65536
  const int pw = P & (HW - 1);
  const int ph = (P >> 6) & (HW - 1);
  const int pn = P >> 12;

  // ================= Phase A: priors via V_WMMA_F32_16X16X4_F32 ============
  for (int t = 0; t < TAPS; ++t) {
    const int f  = t / 9;
    const int ij = t - 9 * f;
    const int hh = ph + ij / 3 - 1;           // pad = 1
    const int ww = pw + ij % 3 - 1;
    const bool inb = (hh >= 0) && (hh < HW) && (ww >= 0) && (ww < HW);

    // A: 16x4 fp32 tile, lane L holds A[M=L%16][K=2*(L>>4)+v]
    const float* xb = x + ((pn * C_IN + f * IN_LEN) * HW + hh) * HW + ww;
    const int d0 = 2 * kgrp;
    v2f a0, a1;
    a0.x = inb ? xb[(d0 + 0) * HW * HW] : 0.0f;
    a0.y = inb ? xb[(d0 + 1) * HW * HW] : 0.0f;
    a1.x = inb ? xb[(d0 + 4) * HW * HW] : 0.0f;
    a1.y = inb ? xb[(d0 + 5) * HW * HW] : 0.0f;

    #pragma unroll
    for (int half = 0; half < 2; ++half) {    // N = 32 (o,u) -> two 16-wide tiles
      const int o = half * 2 + (nn >> 3);
      const int u = nn & 7;
      // weight[o][f][i][j][u][d], B[K=d][N=(o,u)]
      const float* wb = w + (((o * F_IN + f) * 9 + ij) * OUT_LEN + u) * IN_LEN;
      v2f b0, b1;
      b0.x = wb[d0 + 0]; b0.y = wb[d0 + 1];
      b1.x = wb[d0 + 4]; b1.y = wb[d0 + 5];

      v8f c = {};
      c = __builtin_amdgcn_wmma_f32_16x16x4_f32(false, a0, false, b0,
                                                (short)0, c, false, false);
      c = __builtin_amdgcn_wmma_f32_16x16x4_f32(false, a1, false, b1,
                                                (short)0, c, false, false);

      // C tile: VGPR v, lanes 0-15 -> M=v ; lanes 16-31 -> M=v+8 ; N = lane%16
      #pragma unroll
      for (int v = 0; v < 8; ++v) {
        const int pos = v + 8 * kgrp;
        pri_lds[((wave * POS_WAVE + pos) * TAPS + t) * OU + half * 16 + nn] =
            __float2half(c[v]);
      }
    }
  }
  __syncthreads();

  // ================= Phase B: dynamic routing (wave32, lane = (o,u)) =======
  for (int q = 0; q < POS_WAVE; ++q) {
    const int Pq = blockIdx.x * POS_BLK + wave * POS_WAVE + q;

    float pri[TAPS], lg[TAPS];
    #pragma unroll
    for (int t = 0; t < TAPS; ++t) {
      pri[t] = __half2float(pri_lds[((wave * POS_WAVE + q) * TAPS + t) * OU + lane]);
      lg[t]  = 0.0f;
    }

    float outs = 0.0f;
    #pragma unroll
    for (int r = 0; r < 3; ++r) {
      outs = 0.0f;
      #pragma unroll
      for (int t = 0; t < TAPS; ++t) {
        // softmax over o: butterfly across lanes ^8, ^16 (4 o-values / fixed u)
        float b  = lg[t];
        float mx = fmaxf(b, __shfl_xor(b, 8, 32));
        mx       = fmaxf(mx, __shfl_xor(mx, 16, 32));
        float e  = __expf(b - mx);
        float s  = e + __shfl_xor(e, 8, 32);
        s       += __shfl_xor(s, 16, 32);
        outs     = fmaf(e / s, pri[t], outs);      // sum over in-capsules
      }
      if (r != 2) {
        // squash: |v|^2 reduced over u lanes (^1,^2,^4)
        float n2 = outs * outs;
        n2 += __shfl_xor(n2, 1, 32);
        n2 += __shfl_xor(n2, 2, 32);
        n2 += __shfl_xor(n2, 4, 32);
        const float sc = n2 / ((1.0f + n2) * sqrtf(n2 + 1e-12f));
        const float oh = outs * sc;
        // agreement: logits[t] += sum_u pri[t][u] * out_hat[u]
        #pragma unroll
        for (int t = 0; t < TAPS; ++t) {
          float d = pri[t] * oh;
          d += __shfl_xor(d, 1, 32);
          d += __shfl_xor(d, 2, 32);
          d += __shfl_xor(d, 4, 32);
          lg[t] += d;
        }
      }
    }

    // final squash + store: channel = o*8+u = lane
    float n2 = outs * outs;
    n2 += __shfl_xor(n2, 1, 32);
    n2 += __shfl_xor(n2, 2, 32);
    n2 += __shfl_xor(n2, 4, 32);
    const float sc = n2 / ((1.0f + n2) * sqrtf(n2 + 1e-12f));
    const int n  = Pq >> 12;
    const int hw = Pq & 4095;
    out[(n * OU + lane) * (HW * HW) + hw] = outs * sc;
  }
}

extern "C" void kernel_launch(void* const* d_in, const int* in_sizes, int n_in,
                              void* d_out, int out_size, void* d_ws, size_t ws_size,
                              hipStream_t stream) {
  const float* x = (const float*)d_in[0];        // [16,32,64,64] fp32
  const float* w = (const float*)d_in[1];        // [4,4,3,3,8,8] fp32
  float* out = (float*)d_out;                    // [16,32,64,64] fp32

  const int total_pos = N_BATCH * HW * HW;       // 65536
  const int blocks = total_pos / POS_BLK;        // 1024
  const size_t lds_bytes = (size_t)LDS_HALFS * sizeof(__half);  // 147456 B

  capsule_routing_wmma<<<blocks, WAVES * 32, lds_bytes, stream>>>(x, w, out);
}